// QuantumFeatureMap_23115513987345
// MI455X (gfx1250) — compile-verified
//
#include <hip/hip_runtime.h>
#include <stdint.h>

// QuantumFeatureMap on gfx1250 (MI455X).
// Streaming kernel: 160 MiB total traffic -> ~7us roofline at 23.3 TB/s.
// Async global<->LDS staging (CDNA5 ASYNCcnt path) fixes the 40-byte AoS row
// layout; math uses native v_tanh_f32 / v_sin_f32 / v_cos_f32. No WMMA: each
// row has its own 4x4 matrix (per-lane block-diagonal), no shared-matrix
// structure for a wave-wide matrix unit.

#define THREADS 256
#define FEATS 10
#define TILE_FLOATS (THREADS * FEATS)               // 2560 floats = 10240 B
#define VEC2_PER_THREAD (TILE_FLOATS / 2 / THREADS) // 5 float2 per thread
#define STEP_BYTES (THREADS * 8)                    // 2048 B per staging step

__device__ __forceinline__ float fast_tanh(float x) {
#if __has_builtin(__builtin_amdgcn_tanhf)
  return __builtin_amdgcn_tanhf(x);            // v_tanh_f32 (gfx1250)
#elif __has_builtin(__builtin_amdgcn_tanh_f32)
  return __builtin_amdgcn_tanh_f32(x);
#else
  float e = __expf(2.0f * x);
  return 1.0f - 2.0f / (e + 1.0f);             // safe at +-inf
#endif
}

// Z expectations of one matchgate pair.  All complex phases cancel in |.|^2
// except cos(phi1), cos(phi2):
//   p1 = (c u)^2 + (s v)^2 - 2 c s u v cos(phi1)
//   p2 = (s u)^2 + (c v)^2 + 2 c s u v cos(phi2)
__device__ __forceinline__ void pair_z(float ta, float tb, float th,
                                       float ph1, float ph2,
                                       float& zi, float& zj, float& zz) {
  float ca = __cosf(0.5f * ta), sa = __sinf(0.5f * ta);
  float cb = __cosf(0.5f * tb), sb = __sinf(0.5f * tb);
  float c  = __cosf(0.5f * th), s  = __sinf(0.5f * th);
  float cp1 = __cosf(ph1), cp2 = __cosf(ph2);

  float a0 = ca * cb;       // psi0[0]
  float u  = ca * sb;       // psi0[1]
  float v  = sa * cb;       // psi0[2]
  float a3 = sa * sb;       // psi0[3]

  float p0 = a0 * a0;
  float p3 = a3 * a3;
  float cu = c * u, sv = s * v, su = s * u, cv = c * v;
  float cross = 2.0f * (cu * sv);              // 2 c s u v
  float p1 = cu * cu + sv * sv - cross * cp1;
  float p2 = su * su + cv * cv + cross * cp2;

  zi = p0 + p1 - p2 - p3;
  zj = p0 - p1 + p2 - p3;
  zz = p0 - p1 - p2 + p3;
}

__device__ __forceinline__ uint32_t lds_addr_of(const float* p) {
  // Flat LDS addresses: low 32 bits are the workgroup-relative LDS byte
  // offset (ISA 10.2); async ops take that in a VGPR.
  return (uint32_t)(uintptr_t)p;
}

__global__ __launch_bounds__(THREADS)
void qfm_kernel(const float* __restrict__ x, float* __restrict__ out, int nrows) {
  __shared__ float lds_in[TILE_FLOATS];
  __shared__ float lds_out[TILE_FLOATS];

  const int tid = threadIdx.x;
  const int tile_base = (int)blockIdx.x * TILE_FLOATS;   // float index
  const int total = nrows * FEATS;                       // < 2^31
  const bool full_tile = (tile_base + TILE_FLOATS) <= total;  // uniform

  // ---- Stage input tile: async global -> LDS (ASYNCcnt) ----
  // INST_OFFSET is added to BOTH the LDS and global addresses, so all 5
  // steps share one LDS-offset VGPR and one 64-bit address pair.
  if (full_tile) {
    uint32_t lo = lds_addr_of(&lds_in[2 * tid]);
    const void* ga = (const void*)(x + tile_base + 2 * tid);
    asm volatile(
        "global_load_async_to_lds_b64 %0, %1, off\n\t"
        "global_load_async_to_lds_b64 %0, %1, off offset:2048\n\t"
        "global_load_async_to_lds_b64 %0, %1, off offset:4096\n\t"
        "global_load_async_to_lds_b64 %0, %1, off offset:6144\n\t"
        "global_load_async_to_lds_b64 %0, %1, off offset:8192"
        :: "v"(lo), "v"(ga)
        : "memory");
  } else {
    // ragged tail tile (not hit at BATCH=2^21): per-op EXEC-masked guards
#pragma unroll
    for (int k = 0; k < VEC2_PER_THREAD; ++k) {
      int f2 = tid + k * THREADS;
      int g = tile_base + 2 * f2;          // even; total is even
      if (g < total) {
        uint32_t lo = lds_addr_of(&lds_in[2 * f2]);
        const void* ga = (const void*)(x + g);
        asm volatile("global_load_async_to_lds_b64 %0, %1, off"
                     :: "v"(lo), "v"(ga)
                     : "memory");
      }
    }
  }
  asm volatile("s_wait_asynccnt 0x0" ::: "memory");
  __syncthreads();

  // ---- Per-row math from LDS (10-word row stride: bank-conflict-free) ----
  const int row = (int)blockIdx.x * THREADS + tid;
  if (row < nrows) {
    const float* xr = &lds_in[tid * FEATS];
    float t[FEATS];
#pragma unroll
    for (int k = 0; k < FEATS; ++k) t[k] = 0.5f * fast_tanh(xr[k]);

    float z0, z1, zz01, z2, z3, zz23;
    pair_z(t[0], t[1], t[4], t[5], t[6], z0, z1, zz01);
    pair_z(t[2], t[3], t[7], t[8], t[9], z2, z3, zz23);

    float* orow = &lds_out[tid * FEATS];
    orow[0] = z0;       orow[1] = z1;       orow[2] = z2;
    orow[3] = z3;       orow[4] = zz01;
    orow[5] = z0 * z2;  orow[6] = z0 * z3;
    orow[7] = z1 * z2;  orow[8] = z1 * z3;  orow[9] = zz23;
  }
  __syncthreads();

  // ---- Stream results out: async LDS -> global ----
  if (full_tile) {
    uint32_t lo = lds_addr_of(&lds_out[2 * tid]);
    void* ga = (void*)(out + tile_base + 2 * tid);
    asm volatile(
        "global_store_async_from_lds_b64 %0, %1, off\n\t"
        "global_store_async_from_lds_b64 %0, %1, off offset:2048\n\t"
        "global_store_async_from_lds_b64 %0, %1, off offset:4096\n\t"
        "global_store_async_from_lds_b64 %0, %1, off offset:6144\n\t"
        "global_store_async_from_lds_b64 %0, %1, off offset:8192"
        :: "v"(ga), "v"(lo)
        : "memory");
  } else {
#pragma unroll
    for (int k = 0; k < VEC2_PER_THREAD; ++k) {
      int f2 = tid + k * THREADS;
      int g = tile_base + 2 * f2;
      if (g < total) {
        uint32_t lo = lds_addr_of(&lds_out[2 * f2]);
        void* ga = (void*)(out + g);
        asm volatile("global_store_async_from_lds_b64 %0, %1, off"
                     :: "v"(ga), "v"(lo)
                     : "memory");
      }
    }
  }
  asm volatile("s_wait_asynccnt 0x0" ::: "memory");
}

extern "C" void kernel_launch(void* const* d_in, const int* in_sizes, int n_in,
                              void* d_out, int out_size, void* d_ws, size_t ws_size,
                              hipStream_t stream) {
  const float* x = (const float*)d_in[0];
  float* out = (float*)d_out;
  int nrows = in_sizes[0] / FEATS;          // [B,10] row-major
  int blocks = (nrows + THREADS - 1) / THREADS;
  qfm_kernel<<<blocks, THREADS, 0, stream>>>(x, out, nrows);
}